// GlobalFilter_22488448762575
// MI455X (gfx1250) — compile-verified
//
#include <hip/hip_runtime.h>
#include <math.h>

// ---------------------------------------------------------------------------
// GlobalFilter (rfft2 -> per-channel complex weight -> irfft2) for MI455X.
//
// Strategy: DFT-as-GEMM with V_WMMA_F32_16X16X4_F32 (f32 WMMA, no precision
// loss). Channels ride the WMMA N dimension (contiguous in memory -> coalesced
// 64B segments per half-wave). One wave32 = one (batch, 16-channel) tile,
// wave-private LDS staging between stages, zero barriers.
//
// Roofline: 154 MB HBM traffic -> ~6.6us @ 23.3 TB/s; ~6 GFLOP of f32 WMMA is
// negligible. Memory-bound by design.
//
// WMMA f32 16x16x4 operand layout assumptions (per CDNA5 ISA):
//   A (16x4, MxK):  vgpr v, lane l -> M = l%16, K = v + 2*(l/16)
//   B (4x16, KxN):  vgpr v, lane l -> K = v + 2*(l/16), N = l%16
//   C/D (16x16):    vgpr r, lane l -> M = r + 8*(l/16), N = l%16
// ---------------------------------------------------------------------------

typedef __attribute__((ext_vector_type(2))) float v2f;
typedef __attribute__((ext_vector_type(8))) float v8f;

#define DIM   768
#define NPIX  196
#define NCG   48        // 768 / 16 channel groups
#define NB    128

__device__ __forceinline__ v8f wmma4(v2f a, v2f b, v8f c) {
  // 8 args: (neg_a, A, neg_b, B, c_mod, C, reuse_a, reuse_b)
  return __builtin_amdgcn_wmma_f32_16x16x4_f32(
      false, a, false, b, (short)0, c, false, false);
}

// ---------------------------------------------------------------------------
// Constant DFT matrices (7 x 16x16 f32, row-major) written to d_ws:
//  0 GrT[k][j] =  cos(2pi jk/14)                 (k<8,  j<14)  rfft real
//  1 GiT[k][j] = -sin(2pi jk/14)                 (k<8,  j<14)  rfft imag
//  2 Fr [m][a] =  cos(2pi ma/14)                 (m<14, a<14)  fft real
//  3 Fi [m][a] = -sin(2pi ma/14)                 (m<14, a<14)  fft imag
//  4 FiN[m][a] = +sin(2pi ma/14)                 (= -Fi, for complex subtract)
//  5 HrT[n][k] =  alpha_k*cos(2pi kn/14)/196     (n<14, k<8)   irfft fold
//  6 HiT[n][k] = -alpha_k*sin(2pi kn/14)/196     alpha = 1 (k=0,7) else 2
// K-dimension columns beyond the valid range are ZERO so padded garbage rows
// of B operands annihilate.
// ---------------------------------------------------------------------------
__global__ void gf_setup_consts(float* __restrict__ cst) {
  int t = blockIdx.x * blockDim.x + threadIdx.x;
  if (t >= 7 * 256) return;
  int mat = t >> 8;
  int row = (t >> 4) & 15;
  int col = t & 15;
  const float TWO_PI = 6.28318530717958647692f;
  float v = 0.0f;
  float th = TWO_PI * (float)(row * col) / 14.0f;
  if (mat == 0) {
    if (row < 8 && col < 14) v = cosf(th);
  } else if (mat == 1) {
    if (row < 8 && col < 14) v = -sinf(th);
  } else if (mat == 2) {
    if (row < 14 && col < 14) v = cosf(th);
  } else if (mat == 3) {
    if (row < 14 && col < 14) v = -sinf(th);
  } else if (mat == 4) {
    if (row < 14 && col < 14) v = sinf(th);
  } else if (mat == 5) {
    if (row < 14 && col < 8) {
      float alpha = (col == 0 || col == 7) ? 1.0f : 2.0f;
      v = alpha * cosf(th) * (1.0f / 196.0f);
    }
  } else {
    if (row < 14 && col < 8) {
      float alpha = (col == 0 || col == 7) ? 1.0f : 2.0f;
      v = -alpha * sinf(th) * (1.0f / 196.0f);
    }
  }
  cst[t] = v;
}

// LDS cube indexing (per-wave region, all sizes 3584 floats):
//   Y[ri][k(8)][a(14)][c(16)]  and aliased  V[ri][p(14)][k(8)][c(16)]
//   Z[ri][k(8)][m(14)][c(16)]
#define YIDX(ri,k,a,c)  ((((ri)*8 + (k))*14 + (a))*16 + (c))
#define ZIDX(ri,k,m,c)  ((((ri)*8 + (k))*14 + (m))*16 + (c))
#define VIDX(ri,p,k,c)  ((((ri)*14 + (p))*8 + (k))*16 + (c))

__global__ __launch_bounds__(64) void gf_kernel(
    const float* __restrict__ x,      // [128, 196, 768]
    const float* __restrict__ w,      // [14, 8, 768, 2]
    const float* __restrict__ cst,    // 7 x 256 constants
    float* __restrict__ out) {        // [128, 196, 768]
  __shared__ float lds[2 * 2 * 3584];   // 2 waves x (Y/V cube + Z cube)

  const int lane = threadIdx.x & 31;
  const int wv   = threadIdx.x >> 5;
  const int hi   = lane >> 4;           // half-wave (0/1)
  const int ln   = lane & 15;           // lane-in-half == WMMA N / M-low

  float* Ybuf = &lds[wv * 2 * 3584];    // Y cube, later reused as V cube
  float* Zbuf = Ybuf + 3584;            // Z/U cube

  const int tile = blockIdx.x * 2 + wv; // 6144 tiles = 128 b * 48 cgroups
  const int b  = tile / NCG;
  const int c0 = (tile - b * NCG) * 16;

  const float* xb = x + (size_t)b * NPIX * DIM + c0 + ln;
  float*       ob = out + (size_t)b * NPIX * DIM + c0 + ln;
  const float* wb = w + (size_t)(c0 + ln) * 2;

  // ---- load the 7 constant A-operands as register fragments -------------
  v2f A_GrT[4], A_GiT[4], A_Fr[4], A_Fi[4], A_FiN[4], A_HrT[4], A_HiT[4];
#pragma unroll
  for (int q = 0; q < 4; ++q) {
    int off = ln * 16 + 4 * q + 2 * hi;           // A[l%16][4q + v + 2*hi]
    A_GrT[q] = *(const v2f*)(cst + 0 * 256 + off);
    A_GiT[q] = *(const v2f*)(cst + 1 * 256 + off);
    A_Fr [q] = *(const v2f*)(cst + 2 * 256 + off);
    A_Fi [q] = *(const v2f*)(cst + 3 * 256 + off);
    A_FiN[q] = *(const v2f*)(cst + 4 * 256 + off);
    A_HrT[q] = *(const v2f*)(cst + 5 * 256 + off);
    A_HiT[q] = *(const v2f*)(cst + 6 * 256 + off);
  }

  // ---- Stage 1: rfft along j.  Y_a = GrT * X_a  (per a-slice) -----------
  for (int a = 0; a < 14; ++a) {
    v2f Bf[4];
#pragma unroll
    for (int q = 0; q < 4; ++q) {
#pragma unroll
      for (int v = 0; v < 2; ++v) {
        int j = 4 * q + v + 2 * hi;               // B row (K = j)
        if (j > 13) j = 13;                       // clamp; A col is zero there
        Bf[q][v] = xb[(a * 14 + j) * DIM];        // coalesced 64B/half-wave
      }
    }
    v8f Cr = {}, Ci = {};
#pragma unroll
    for (int q = 0; q < 4; ++q) {
      Cr = wmma4(A_GrT[q], Bf[q], Cr);
      Ci = wmma4(A_GiT[q], Bf[q], Ci);
    }
    if (hi == 0) {                                // rows k = 0..7 live here
#pragma unroll
      for (int r = 0; r < 8; ++r) {
        Ybuf[YIDX(0, r, a, ln)] = Cr[r];
        Ybuf[YIDX(1, r, a, ln)] = Ci[r];
      }
    }
  }

  // ---- Stage 2: fft along a, then complex weight multiply ---------------
  for (int k = 0; k < 8; ++k) {
    v2f Br[4], Bi[4];
#pragma unroll
    for (int q = 0; q < 4; ++q) {
#pragma unroll
      for (int v = 0; v < 2; ++v) {
        int aa = 4 * q + v + 2 * hi;
        if (aa > 13) aa = 13;                     // A col zero beyond 13
        Br[q][v] = Ybuf[YIDX(0, k, aa, ln)];
        Bi[q][v] = Ybuf[YIDX(1, k, aa, ln)];
      }
    }
    v8f Zr = {}, Zi = {};
#pragma unroll
    for (int q = 0; q < 4; ++q) {
      Zr = wmma4(A_Fr [q], Br[q], Zr);            // Fr*Yr
      Zr = wmma4(A_FiN[q], Bi[q], Zr);            // - Fi*Yi
      Zi = wmma4(A_Fi [q], Br[q], Zi);            // Fi*Yr
      Zi = wmma4(A_Fr [q], Bi[q], Zi);            // + Fr*Yi
    }
    // per-lane complex weight multiply on accumulator registers
#pragma unroll
    for (int r = 0; r < 8; ++r) {
      int m = r + 8 * hi;
      int me = (m > 13) ? 13 : m;
      v2f wc = *(const v2f*)(wb + (size_t)((me * 8 + k) * DIM) * 2);
      float ur = wc[0] * Zr[r] - wc[1] * Zi[r];
      float ui = wc[0] * Zi[r] + wc[1] * Zr[r];
      if (m < 14) {
        Zbuf[ZIDX(0, k, m, ln)] = ur;
        Zbuf[ZIDX(1, k, m, ln)] = ui;
      }
    }
  }

  // ---- Stage 3: inverse fft along m.  V_k = conj(F) * U_k ---------------
  // (Y cube is dead; reuse as V cube. LDS ops are in-order within a wave.)
  for (int k = 0; k < 8; ++k) {
    v2f Br[4], Bi[4];
#pragma unroll
    for (int q = 0; q < 4; ++q) {
#pragma unroll
      for (int v = 0; v < 2; ++v) {
        int m = 4 * q + v + 2 * hi;
        if (m > 13) m = 13;
        Br[q][v] = Zbuf[ZIDX(0, k, m, ln)];
        Bi[q][v] = Zbuf[ZIDX(1, k, m, ln)];
      }
    }
    v8f Vr = {}, Vi = {};
#pragma unroll
    for (int q = 0; q < 4; ++q) {
      Vr = wmma4(A_Fr [q], Br[q], Vr);            // Fr*Ur
      Vr = wmma4(A_Fi [q], Bi[q], Vr);            // + Fi*Ui   (conj(F))
      Vi = wmma4(A_Fr [q], Bi[q], Vi);            // Fr*Ui
      Vi = wmma4(A_FiN[q], Br[q], Vi);            // - Fi*Ur
    }
#pragma unroll
    for (int r = 0; r < 8; ++r) {
      int p = r + 8 * hi;
      if (p < 14) {
        Ybuf[VIDX(0, p, k, ln)] = Vr[r];
        Ybuf[VIDX(1, p, k, ln)] = Vi[r];
      }
    }
  }

  // ---- Stage 4: Hermitian irfft along last axis, write output -----------
  for (int p = 0; p < 14; ++p) {
    v2f Br[4], Bi[4];
#pragma unroll
    for (int q = 0; q < 4; ++q) {
#pragma unroll
      for (int v = 0; v < 2; ++v) {
        int k = 4 * q + v + 2 * hi;
        if (k > 7) k = 7;                         // A col zero beyond 7
        Br[q][v] = Ybuf[VIDX(0, p, k, ln)];
        Bi[q][v] = Ybuf[VIDX(1, p, k, ln)];
      }
    }
    v8f O = {};
#pragma unroll
    for (int q = 0; q < 4; ++q) {
      O = wmma4(A_HrT[q], Br[q], O);
      O = wmma4(A_HiT[q], Bi[q], O);
    }
#pragma unroll
    for (int r = 0; r < 8; ++r) {
      int n = r + 8 * hi;
      if (n < 14) ob[(p * 14 + n) * DIM] = O[r];  // coalesced 64B/half-wave
    }
  }
}

extern "C" void kernel_launch(void* const* d_in, const int* in_sizes, int n_in,
                              void* d_out, int out_size, void* d_ws, size_t ws_size,
                              hipStream_t stream) {
  const float* x = (const float*)d_in[0];          // [128,196,768] f32
  const float* w = (const float*)d_in[1];          // [14,8,768,2]  f32
  float* out = (float*)d_out;                      // [128,196,768] f32
  float* cst = (float*)d_ws;                       // needs 7*256*4 = 7168 B

  gf_setup_consts<<<7, 256, 0, stream>>>(cst);
  // 6144 tiles (128 batches x 48 channel-groups), 2 waves per block
  gf_kernel<<<3072, 64, 0, stream>>>(x, w, cst, out);
}